// ObjectAlignmentNet_44263932952625
// MI455X (gfx1250) — compile-verified
//
#include <hip/hip_runtime.h>

typedef float v2f __attribute__((ext_vector_type(2)));
typedef float v8f __attribute__((ext_vector_type(8)));

#define B_TOT   128
#define WD      300
#define ID      2048
#define K_TOT   (WD * ID)      // 614400
#define KT      512            // k per tile (divides ID)
#define TPB_T   8              // tiles per block
#define KCHUNKS 150            // KCHUNKS * TPB_T * KT == K_TOT
#define PADROW  (KT + 4)       // LDS row pitch: 516 % 64 == 4 -> conflict-free

// Seed output with the bias; main kernel accumulates with atomics on top.
__global__ __launch_bounds__(256)
void oan_init(const float* __restrict__ align_b, float* __restrict__ out) {
    int i = threadIdx.x;             // out_size == 256 == blockDim
    out[i] = align_b[i & 1];
}

__global__ __launch_bounds__(256)
void oan_main(const float* __restrict__ ims,
              const int*   __restrict__ words,
              const float* __restrict__ embed_w,
              const float* __restrict__ align_w,
              const float* __restrict__ drop_mask,
              float*       __restrict__ out) {
    __shared__ float lds_wv[16 * WD];        // 16 embedding rows
    __shared__ float lds_feat[16 * PADROW];  // 16 x 512 feat slab (padded)
    __shared__ float lds_bw[2 * KT];         // align_w slice
    __shared__ float lds_red[8 * 32];        // cross-wave reduction

    const int tid  = threadIdx.x;
    const int lane = tid & 31;
    const int wave = tid >> 5;
    const int b0   = blockIdx.y * 16;

    // Stage wv[r][:] = embed_w[words[b0+r]][:]  (once per block; off hot path)
    for (int r = 0; r < 16; ++r) {
        const int word = words[b0 + r];
        const float* erow = embed_w + (size_t)word * WD;
        for (int wcol = tid; wcol < WD; wcol += 256)
            lds_wv[r * WD + wcol] = erow[wcol];
    }

    v8f c = {};
    const int mrow  = lane & 15;       // M (A) or N (B) index
    const int hi    = lane >> 4;       // half-wave -> K pair select
    const int kwave = wave * 64;       // this wave's k slice inside tile

    // Streaming layout: threads 0-127 -> even row of a pair (float4 cols 0..508),
    // threads 128-255 -> odd row. All staging traffic is b128.
    const int rowIn = tid >> 7;        // 0 or 1
    const int col4  = (tid & 127) * 4; // 0..508 step 4

    for (int tt = 0; tt < TPB_T; ++tt) {
        __syncthreads();               // protect LDS reuse (and wv on 1st iter)
        const int kc    = (blockIdx.x * TPB_T + tt) * KT;
        const int w     = kc >> 11;          // constant within tile (KT | ID)
        const int iBase = kc & (ID - 1);

        // Stage align_w[0..1, kc : kc+KT): exactly one b128 per thread.
        *(float4*)&lds_bw[rowIn * KT + col4] =
            *(const float4*)(align_w + (size_t)rowIn * K_TOT + kc + col4);

        // Stream mask (the 315 MB tensor) with b128 loads, fuse wv*ims*mask -> LDS
        #pragma unroll
        for (int rp = 0; rp < 8; ++rp) {
            const int r = rp * 2 + rowIn;
            const float4 m4 = *(const float4*)(drop_mask + (size_t)(b0 + r) * K_TOT + kc + col4);
            const float4 i4 = *(const float4*)(ims       + (size_t)(b0 + r) * ID    + iBase + col4);
            const float wvr = lds_wv[r * WD + w];
            float4 f4;
            f4.x = wvr * i4.x * m4.x;
            f4.y = wvr * i4.y * m4.y;
            f4.z = wvr * i4.z * m4.z;
            f4.w = wvr * i4.w * m4.w;
            *(float4*)&lds_feat[r * PADROW + col4] = f4;
        }
        // Warm L2/WGP$ for the next tile (same rows, +KT floats). Speculative;
        // overshoot past the buffer end on the final tile is dropped harmlessly.
        #pragma unroll
        for (int rp = 0; rp < 8; ++rp) {
            const int r = rp * 2 + rowIn;
            __builtin_prefetch(drop_mask + (size_t)(b0 + r) * K_TOT + kc + KT + col4);
        }
        __syncthreads();

        // WMMA accumulate: C[16x16] += A(16x4) x B(4x16) over this wave's k slice
        #pragma unroll
        for (int step = 0; step < 16; ++step) {
            const int ko = kwave + step * 4;
            v2f a, b;
            const float2 av = *(const float2*)&lds_feat[mrow * PADROW + ko + 2 * hi];
            a.x = av.x; a.y = av.y;
            if (mrow < 2) {                  // only 2 real output columns
                const float2 bv = *(const float2*)&lds_bw[mrow * KT + ko + 2 * hi];
                b.x = bv.x; b.y = bv.y;
            } else {
                b.x = 0.0f; b.y = 0.0f;
            }
            c = __builtin_amdgcn_wmma_f32_16x16x4_f32(
                    false, a, false, b, (short)0, c, false, false);
        }
    }

    __syncthreads();
    // C layout: VGPR r, lanes 0-15 -> (M=r, N=lane), lanes 16-31 -> (M=r+8, N=lane-16).
    // Only N in {0,1} is meaningful -> lanes 0,1,16,17 hold everything.
    if ((lane & 0x0E) == 0) {
        int n  = lane & 1;
        int mh = (lane >> 4) * 8;
        #pragma unroll
        for (int r = 0; r < 8; ++r)
            lds_red[wave * 32 + (mh + r) * 2 + n] = c[r];
    }
    __syncthreads();
    if (tid < 32) {
        float s = 0.0f;
        #pragma unroll
        for (int wv = 0; wv < 8; ++wv) s += lds_red[wv * 32 + tid];
        int M = tid >> 1, n = tid & 1;
        atomicAdd(&out[(b0 + M) * 2 + n], s);
    }
}

extern "C" void kernel_launch(void* const* d_in, const int* in_sizes, int n_in,
                              void* d_out, int out_size, void* d_ws, size_t ws_size,
                              hipStream_t stream) {
    const float* ims       = (const float*)d_in[0];
    const int*   words     = (const int*)  d_in[1];
    const float* embed_w   = (const float*)d_in[2];
    const float* align_w   = (const float*)d_in[3];
    const float* align_b   = (const float*)d_in[4];
    const float* drop_mask = (const float*)d_in[5];
    float* out = (float*)d_out;

    oan_init<<<1, 256, 0, stream>>>(align_b, out);
    dim3 grid(KCHUNKS, B_TOT / 16);
    oan_main<<<grid, 256, 0, stream>>>(ims, words, embed_w, align_w, drop_mask, out);
}